// HypernymVisual_cos_10213432230334
// MI455X (gfx1250) — compile-verified
//
#include <hip/hip_runtime.h>

#define Bq 2048
#define Dq 4096
#define Eq 512
#define Nq 8192

typedef __attribute__((ext_vector_type(2))) float v2f;
typedef __attribute__((ext_vector_type(8))) float v8f;
typedef __attribute__((ext_vector_type(4))) int v4i;

#define AS1 __attribute__((address_space(1)))
#define AS3 __attribute__((address_space(3)))

// CDNA5 async global->LDS copy path (ASYNCcnt / s_wait_asynccnt), guarded so
// the file still compiles if the toolchain lacks the builtins.
#if defined(__gfx1250__) && __has_builtin(__builtin_amdgcn_global_load_async_to_lds_b128)
#define USE_ASYNC_COPY 1
#else
#define USE_ASYNC_COPY 0
#endif

#if USE_ASYNC_COPY
#if __has_builtin(__builtin_amdgcn_s_wait_asynccnt)
#define WAIT_ASYNC(n) __builtin_amdgcn_s_wait_asynccnt(n)
#else
#define WAIT_ASYNC(n) asm volatile("s_wait_asynccnt %0" ::"i"(n))
#endif
__device__ __forceinline__ void async_copy16(const float* g, float* l) {
  // prototype (from hipcc diagnostic): param0 = AS1 v4i* (global src),
  // param1 = AS3 v4i* (LDS dst), then imm offset, imm cpol.
  __builtin_amdgcn_global_load_async_to_lds_b128((AS1 v4i*)g, (AS3 v4i*)l, 0,
                                                 0);
}
#endif

// LDS row stride: 36 floats. 16B-aligned segments (36*4 % 16 == 0) for B128
// LDS writes, and conflict-free fragment reads (m*36 mod 64 -> 16 distinct
// banks; the lane16-31 half reads the +2 bank class, disjoint).
#define LDSS 36

// ---------------------------------------------------------------------------
// Kernel 1: BatchNorm stats over batch dim -> fused scale/shift per feature
// ---------------------------------------------------------------------------
__global__ void bn_stats_kernel(const float* __restrict__ vf,
                                const float* __restrict__ gamma,
                                const float* __restrict__ beta,
                                float* __restrict__ scale,
                                float* __restrict__ shift) {
  int d = blockIdx.x * blockDim.x + threadIdx.x;
  float s = 0.f, sq = 0.f;
  for (int r = 0; r < Bq; ++r) {
    float x = vf[(size_t)r * Dq + d];  // coalesced across threads
    s += x;
    sq += x * x;
  }
  float mean = s * (1.0f / (float)Bq);
  float var = sq * (1.0f / (float)Bq) - mean * mean;
  float sc = gamma[d] * rsqrtf(var + 1e-5f);
  scale[d] = sc;
  shift[d] = beta[d] - mean * sc;
}

// ---------------------------------------------------------------------------
// Kernel 2: vfn = vf*scale + shift (materialized for GEMM1), plus row cosine
// ---------------------------------------------------------------------------
__global__ void vfn_pe_kernel(const float* __restrict__ vf,
                              const float* __restrict__ p_wfs,
                              const float* __restrict__ scale,
                              const float* __restrict__ shift,
                              float* __restrict__ vfn,
                              float* __restrict__ pe) {
  __shared__ float r0[256], r1[256], r2[256];
  const int v = blockIdx.x;
  const int t = threadIdx.x;
  const float* vr = vf + (size_t)v * Dq;
  const float* pr = p_wfs + (size_t)v * Dq;
  float* outr = vfn + (size_t)v * Dq;
  float svv = 0.f, spp = 0.f, svp = 0.f;
  for (int j = t; j < Dq; j += 256) {
    float x = vr[j] * scale[j] + shift[j];
    float p = pr[j];
    outr[j] = x;
    svv += x * x;
    spp += p * p;
    svp += x * p;
  }
  r0[t] = svv;
  r1[t] = spp;
  r2[t] = svp;
  __syncthreads();
  for (int off = 128; off > 0; off >>= 1) {
    if (t < off) {
      r0[t] += r0[t + off];
      r1[t] += r1[t + off];
      r2[t] += r2[t + off];
    }
    __syncthreads();
  }
  if (t == 0) {
    float nv = fmaxf(sqrtf(r0[0]), 1e-8f);
    float np = fmaxf(sqrtf(r1[0]), 1e-8f);
    pe[v] = r2[0] / (nv * np);
  }
}

// ---------------------------------------------------------------------------
// WMMA fp32 GEMM, NT layout: C[M,N] = A[M,K] * Bm[N,K]^T
// Block tile 64x64, BK=32. 8 waves: 4 along M x 2 along N; each wave owns two
// 16x16 f32 accumulators. Copy pipeline:
//   async path : GLOBAL_LOAD_ASYNC_TO_LDS_B128 ping-pong buffers, s_wait_asynccnt
//   sync  path : register double-buffer (next chunk's global loads issued
//                before the WMMA loop; wait lands at next iteration's ds_store)
// V_WMMA_F32_16X16X4_F32 fragments:
//   A frag (2 VGPR): a[r] = A[lane%16][k0 + r + 2*(lane>>4)]
//   B frag (2 VGPR): b[r] = Bm[n = lane%16][k0 + r + 2*(lane>>4)]
//   C/D  (8 VGPR):   acc[i] = C[i + 8*(lane>>4)][lane%16]
// ---------------------------------------------------------------------------
template <bool BIAS_RELU>
__global__ void gemm_nt_wmma_kernel(const float* __restrict__ A,
                                    const float* __restrict__ Bm,
                                    const float* __restrict__ bias,
                                    float* __restrict__ C,
                                    int K, int ldc) {
#if USE_ASYNC_COPY
  __shared__ float As[2][64][LDSS];
  __shared__ float Bs[2][64][LDSS];
#else
  __shared__ float As[1][64][LDSS];
  __shared__ float Bs[1][64][LDSS];
#endif

  const int t = threadIdx.x;
  const int lane = t & 31;
  const int wave = t >> 5;
  const int wm = wave & 3;   // wave row (0..3) -> 16 rows each
  const int wn = wave >> 2;  // wave col (0..1) -> 32 cols each
  const int row0 = blockIdx.y * 64;
  const int col0 = blockIdx.x * 64;
  const int mn = lane & 15;  // M index for A frag, N index for B frag
  const int h = lane >> 4;   // half select (K split)

  v8f acc0 = {};
  v8f acc1 = {};

  // cooperative fill mapping: 256 threads, 64 rows x 32 floats, 8 floats/thread
  const int lr = t >> 2;
  const int lc = (t & 3) * 8;
  const float* ga = A + (size_t)(row0 + lr) * K + lc;
  const float* gb = Bm + (size_t)(col0 + lr) * K + lc;

#if USE_ASYNC_COPY
  // prime buffer 0 with chunk 0
  async_copy16(ga, &As[0][lr][lc]);
  async_copy16(ga + 4, &As[0][lr][lc + 4]);
  async_copy16(gb, &Bs[0][lr][lc]);
  async_copy16(gb + 4, &Bs[0][lr][lc + 4]);
  int cur = 0;
#else
  float4 ra0 = *(const float4*)(ga);
  float4 ra1 = *(const float4*)(ga + 4);
  float4 rb0 = *(const float4*)(gb);
  float4 rb1 = *(const float4*)(gb + 4);
  const int cur = 0;
#endif

  for (int k0 = 0; k0 < K; k0 += 32) {
    const int kn = k0 + 32;
#if USE_ASYNC_COPY
    // issue next chunk into the other buffer (its last consumer passed the
    // barrier at the end of the previous iteration)
    if (kn < K) {
      const int nxt = cur ^ 1;
      async_copy16(ga + kn, &As[nxt][lr][lc]);
      async_copy16(ga + kn + 4, &As[nxt][lr][lc + 4]);
      async_copy16(gb + kn, &Bs[nxt][lr][lc]);
      async_copy16(gb + kn + 4, &Bs[nxt][lr][lc + 4]);
      WAIT_ASYNC(4);  // current buffer's 4 ops retired; next 4 may be in flight
    } else {
      WAIT_ASYNC(0);
    }
    __syncthreads();  // all waves' current-buffer data visible
#else
    // store the prefetched chunk (previous compute finished at the barrier
    // that ended the last iteration)
    As[0][lr][lc + 0] = ra0.x; As[0][lr][lc + 1] = ra0.y;
    As[0][lr][lc + 2] = ra0.z; As[0][lr][lc + 3] = ra0.w;
    As[0][lr][lc + 4] = ra1.x; As[0][lr][lc + 5] = ra1.y;
    As[0][lr][lc + 6] = ra1.z; As[0][lr][lc + 7] = ra1.w;
    Bs[0][lr][lc + 0] = rb0.x; Bs[0][lr][lc + 1] = rb0.y;
    Bs[0][lr][lc + 2] = rb0.z; Bs[0][lr][lc + 3] = rb0.w;
    Bs[0][lr][lc + 4] = rb1.x; Bs[0][lr][lc + 5] = rb1.y;
    Bs[0][lr][lc + 6] = rb1.z; Bs[0][lr][lc + 7] = rb1.w;
    __syncthreads();
    if (kn < K) {  // issue next chunk's loads; they retire under the WMMAs
      ra0 = *(const float4*)(ga + kn);
      ra1 = *(const float4*)(ga + kn + 4);
      rb0 = *(const float4*)(gb + kn);
      rb1 = *(const float4*)(gb + kn + 4);
    }
#endif

    const float* ar = &As[cur][wm * 16 + mn][0];
    const float* br0 = &Bs[cur][wn * 32 + mn][0];
    const float* br1 = &Bs[cur][wn * 32 + 16 + mn][0];
#pragma unroll
    for (int kk = 0; kk < 32; kk += 4) {
      const int kb = kk + 2 * h;
      v2f af;  af.x = ar[kb];   af.y = ar[kb + 1];
      v2f bf0; bf0.x = br0[kb]; bf0.y = br0[kb + 1];
      v2f bf1; bf1.x = br1[kb]; bf1.y = br1[kb + 1];
      acc0 = __builtin_amdgcn_wmma_f32_16x16x4_f32(false, af, false, bf0,
                                                   (short)0, acc0, false, false);
      acc1 = __builtin_amdgcn_wmma_f32_16x16x4_f32(false, af, false, bf1,
                                                   (short)0, acc1, false, false);
    }
    __syncthreads();  // everyone done with this buffer before it is refilled
#if USE_ASYNC_COPY
    cur ^= 1;
#endif
  }

  const int gn0 = col0 + wn * 32 + mn;
  const int gn1 = gn0 + 16;
#pragma unroll
  for (int i = 0; i < 8; ++i) {
    const int gm = row0 + wm * 16 + i + 8 * h;
    float v0 = acc0[i];
    float v1 = acc1[i];
    if (BIAS_RELU) {
      v0 = fmaxf(v0 + bias[gn0], 0.f);
      v1 = fmaxf(v1 + bias[gn1], 0.f);
    }
    C[(size_t)gm * ldc + gn0] = v0;
    C[(size_t)gm * ldc + gn1] = v1;
  }
}

// ---------------------------------------------------------------------------
// Kernel 4: row-normalize emb (in place, rows 0..B-1) and n_wfs -> n_norm
// ---------------------------------------------------------------------------
__global__ void norm_rows_kernel(float* __restrict__ emb,
                                 const float* __restrict__ n_wfs,
                                 float* __restrict__ n_norm) {
  __shared__ float red[256];
  const int row = blockIdx.x;
  const int t = threadIdx.x;
  const float* src;
  float* dst;
  if (row < Bq) {
    src = emb + (size_t)row * Eq;
    dst = emb + (size_t)row * Eq;
  } else {
    src = n_wfs + (size_t)(row - Bq) * Eq;
    dst = n_norm + (size_t)(row - Bq) * Eq;
  }
  float x0 = src[t];
  float x1 = src[t + 256];
  red[t] = x0 * x0 + x1 * x1;
  __syncthreads();
  for (int off = 128; off > 0; off >>= 1) {
    if (t < off) red[t] += red[t + off];
    __syncthreads();
  }
  float inv = 1.0f / fmaxf(sqrtf(red[0]), 1e-8f);
  dst[t] = x0 * inv;
  dst[t + 256] = x1 * inv;
}

// ---------------------------------------------------------------------------
// Kernel 6: p_e broadcast (float4 stores): out[v*N + n] = pe[v], N = 1<<13
// ---------------------------------------------------------------------------
__global__ void pe_broadcast_kernel(const float* __restrict__ pe,
                                    float* __restrict__ out) {
  size_t q = (size_t)blockIdx.x * 256 + threadIdx.x;  // float4 index
  float p = pe[q >> 11];                              // (q*4) >> 13
  float4 v = make_float4(p, p, p, p);
  ((float4*)out)[q] = v;
}

// ---------------------------------------------------------------------------
extern "C" void kernel_launch(void* const* d_in, const int* in_sizes, int n_in,
                              void* d_out, int out_size, void* d_ws,
                              size_t ws_size, hipStream_t stream) {
  (void)in_sizes; (void)n_in; (void)out_size; (void)ws_size;
  const float* vf = (const float*)d_in[0];
  const float* p_wfs = (const float*)d_in[1];
  const float* n_wfs = (const float*)d_in[2];
  const float* gamma = (const float*)d_in[3];
  const float* beta = (const float*)d_in[4];
  const float* W = (const float*)d_in[5];
  const float* bvec = (const float*)d_in[6];
  float* out = (float*)d_out;

  // workspace layout (floats): scale[D] shift[D] pe[B] vfn[B*D] emb[B*E] n_norm[N*E]
  float* ws = (float*)d_ws;
  float* scale = ws;
  float* shift = scale + Dq;
  float* pe = shift + Dq;
  float* vfn = pe + Bq;
  float* emb = vfn + (size_t)Bq * Dq;
  float* n_norm = emb + (size_t)Bq * Eq;

  // 1) BN stats -> fused scale/shift
  bn_stats_kernel<<<Dq / 256, 256, 0, stream>>>(vf, gamma, beta, scale, shift);

  // 2) vfn materialization + p_e cosine
  vfn_pe_kernel<<<Bq, 256, 0, stream>>>(vf, p_wfs, scale, shift, vfn, pe);

  // 3) emb = relu(vfn @ W^T + b)   [2048 x 512, K=4096]
  gemm_nt_wmma_kernel<true><<<dim3(Eq / 64, Bq / 64), 256, 0, stream>>>(
      vfn, W, bvec, emb, Dq, Eq);

  // 4) row-normalize emb (in place) and n_wfs -> n_norm
  norm_rows_kernel<<<Bq + Nq, 256, 0, stream>>>(emb, n_wfs, n_norm);

  // 5) n_e = emb_norm @ n_norm^T   [2048 x 8192, K=512] -> d_out[0 : B*N)
  gemm_nt_wmma_kernel<false><<<dim3(Nq / 64, Bq / 64), 256, 0, stream>>>(
      emb, n_norm, nullptr, out, Eq, Nq);

  // 6) p_e broadcast -> d_out[B*N : 2*B*N)
  pe_broadcast_kernel<<<((size_t)Bq * Nq / 4) / 256, 256, 0, stream>>>(
      pe, out + (size_t)Bq * Nq);
}